// ConsRec4RGI_90658169684590
// MI455X (gfx1250) — compile-verified
//
#include <hip/hip_runtime.h>
#include <hip/hip_bf16.h>
#include <math.h>

typedef __attribute__((ext_vector_type(2))) float v2f;
typedef __attribute__((ext_vector_type(8))) float v8f;

#define U_CNT   200000
#define I_CNT   100000
#define G_CNT   10000
#define D_DIM   64
#define B_CNT   8192
#define EHG_CNT 3000000
#define EGI_CNT 2000000
#define NHG_ROWS (U_CNT + G_CNT)   // 210000
#define NGI_ROWS (G_CNT + I_CNT)   // 110000
#define KCHUNK   640               // K-split chunk (multiple of 4)
#define NKCHUNKS ((G_CNT + KCHUNK - 1) / KCHUNK)   // 16

// ---------------------------------------------------------------------------
// concat: dst0 (and optionally dst1) = [a ; b], element-parallel
// ---------------------------------------------------------------------------
__global__ void concat2_kernel(float* __restrict__ dst0, float* __restrict__ dst1,
                               const float* __restrict__ a, size_t na,
                               const float* __restrict__ b, size_t nb) {
  size_t t = (size_t)blockIdx.x * blockDim.x + threadIdx.x;
  size_t n = na + nb;
  if (t >= n) return;
  float v = (t < na) ? a[t] : b[t - na];
  dst0[t] = v;
  if (dst1) dst1[t] = v;
}

// ---------------------------------------------------------------------------
// acc += src
// ---------------------------------------------------------------------------
__global__ void add_inplace_kernel(float* __restrict__ acc,
                                   const float* __restrict__ src, size_t n) {
  size_t t = (size_t)blockIdx.x * blockDim.x + threadIdx.x;
  if (t < n) acc[t] += src[t];
}

// ---------------------------------------------------------------------------
// Edge-parallel SpMM: y[rows[e]] += vals[e] * x[cols[e]]   (rows of D=64)
// One wave32 per edge; each lane handles 2 consecutive dims (float2).
// x/y are L2-resident (<= 54 MB on a 192 MB L2), edge arrays stream from HBM.
// ---------------------------------------------------------------------------
__global__ void spmm_atomic_kernel(const int* __restrict__ rows,
                                   const int* __restrict__ cols,
                                   const float* __restrict__ vals,
                                   const float* __restrict__ x,
                                   float* __restrict__ y, int E) {
  int lane  = threadIdx.x & 31;
  int wave  = (int)((blockIdx.x * blockDim.x + threadIdx.x) >> 5);
  int nwave = (int)((gridDim.x * blockDim.x) >> 5);
  for (int e = wave; e < E; e += nwave) {
    int   r = rows[e];
    int   c = cols[e];
    float v = vals[e];
    const float2 xv = *reinterpret_cast<const float2*>(x + (size_t)c * D_DIM + lane * 2);
    float* yp = y + (size_t)r * D_DIM + lane * 2;
    atomicAdd(yp + 0, v * xv.x);
    atomicAdd(yp + 1, v * xv.y);
  }
}

// ---------------------------------------------------------------------------
// Dense GEMM C = A(G x G) * B(G x 64) with V_WMMA_F32_16X16X4_F32.
// K-split for memory-level parallelism: grid = (row-slabs, K-chunks).
// 625 slabs x 16 chunks = 10000 waves, enough outstanding loads to pull
// 400 MB of A at HBM rate (23.3 TB/s => ~17 us floor). Each wave computes a
// partial 16x64 slab (4x v8f accumulators) over its 640-wide K range and
// commits with global_atomic_add_f32 into the pre-zeroed C.
// A-operand layout (ISA 7.12.2, 32-bit A 16x4): lanes 0-15 -> K={k,k+1},
// lanes 16-31 -> K={k+2,k+3}; M = lane&15. B mirrored over K with N = lane&15.
// C/D layout: VGPR j -> M = j (+8 for lanes 16-31), N = lane&15.
// ---------------------------------------------------------------------------
__global__ void gg_wmma_kernel(const float* __restrict__ A,
                               const float* __restrict__ B,
                               float* __restrict__ C) {
  const int G = G_CNT;
  int wavesPerBlock = blockDim.x >> 5;
  int slab = blockIdx.x * wavesPerBlock + (threadIdx.x >> 5);
  int lane = threadIdx.x & 31;
  int nSlabs = G / 16;                 // 625
  if (slab >= nSlabs) return;

  int k0   = blockIdx.y * KCHUNK;
  int kEnd = k0 + KCHUNK; if (kEnd > G) kEnd = G;

  int r0    = slab * 16;
  int mrow  = lane & 15;
  int khalf = lane >> 4;               // 0 -> K={k,k+1}, 1 -> K={k+2,k+3}

  v8f acc[4];
  acc[0] = v8f{}; acc[1] = v8f{}; acc[2] = v8f{}; acc[3] = v8f{};

  const float* arow = A + (size_t)(r0 + mrow) * G;

  for (int k = k0; k < kEnd; k += 4) {
    if ((k & 255) == 0) {
      // pull upcoming A bytes toward the caches (global_prefetch_b8)
      __builtin_prefetch(arow + k + 256, 0, 1);
    }
    int ka = k + khalf * 2;
    v2f a;
    a.x = arow[ka + 0];
    a.y = arow[ka + 1];
    const float* brow0 = B + (size_t)(ka + 0) * D_DIM;
    const float* brow1 = B + (size_t)(ka + 1) * D_DIM;
#pragma unroll
    for (int t = 0; t < 4; ++t) {
      v2f b;
      b.x = brow0[t * 16 + mrow];
      b.y = brow1[t * 16 + mrow];
      acc[t] = __builtin_amdgcn_wmma_f32_16x16x4_f32(
          /*neg_a=*/false, a, /*neg_b=*/false, b,
          /*c_mod=*/(short)0, acc[t], /*reuse_a=*/false, /*reuse_b=*/false);
    }
  }

  // commit partial slab: atomic accumulate across K-chunks (C pre-zeroed)
  int n   = lane & 15;
  int mhi = (lane >> 4) * 8;
#pragma unroll
  for (int t = 0; t < 4; ++t) {
#pragma unroll
    for (int j = 0; j < 8; ++j) {
      atomicAdd(&C[(size_t)(r0 + mhi + j) * D_DIM + t * 16 + n], acc[t][j]);
    }
  }
}

// ---------------------------------------------------------------------------
// Per-group sigmoid gates + combine:
//   hc = sig(hg.w_h + b_h), lc = sig(gi.w_l + b_l), oc = sig(gg.w_o + b_o)
//   groups_final = hc*hg + lc*gi + oc*gg        (hg = hg_acc[U+g]/4)
// One wave32 per group; 2 dims/lane; __shfl_xor wave reduction.
// ---------------------------------------------------------------------------
__device__ __forceinline__ float wave_sum(float v) {
#pragma unroll
  for (int off = 16; off > 0; off >>= 1) v += __shfl_xor(v, off, 32);
  return v;
}

__global__ void coef_combine_kernel(const float* __restrict__ hg_acc,
                                    const float* __restrict__ gi_out,
                                    const float* __restrict__ gg_out,
                                    const float* __restrict__ hw, const float* __restrict__ hb,
                                    const float* __restrict__ lw, const float* __restrict__ lb,
                                    const float* __restrict__ ow, const float* __restrict__ ob,
                                    float* __restrict__ grp_final) {
  int wavesPerBlock = blockDim.x >> 5;
  int g    = blockIdx.x * wavesPerBlock + (threadIdx.x >> 5);
  int lane = threadIdx.x & 31;
  if (g >= G_CNT) return;

  int d = lane * 2;
  float2 hg = *reinterpret_cast<const float2*>(hg_acc + (size_t)(U_CNT + g) * D_DIM + d);
  hg.x *= 0.25f; hg.y *= 0.25f;
  float2 gi = *reinterpret_cast<const float2*>(gi_out + (size_t)g * D_DIM + d);
  float2 gg = *reinterpret_cast<const float2*>(gg_out + (size_t)g * D_DIM + d);
  float2 wh = *reinterpret_cast<const float2*>(hw + d);
  float2 wl = *reinterpret_cast<const float2*>(lw + d);
  float2 wo = *reinterpret_cast<const float2*>(ow + d);

  float ph = wave_sum(hg.x * wh.x + hg.y * wh.y);
  float pl = wave_sum(gi.x * wl.x + gi.y * wl.y);
  float po = wave_sum(gg.x * wo.x + gg.y * wo.y);

  float hc = 1.0f / (1.0f + __expf(-(ph + hb[0])));
  float lc = 1.0f / (1.0f + __expf(-(pl + lb[0])));
  float oc = 1.0f / (1.0f + __expf(-(po + ob[0])));

  float2 outv;
  outv.x = hc * hg.x + lc * gi.x + oc * gg.x;
  outv.y = hc * hg.y + lc * gi.y + oc * gg.y;
  *reinterpret_cast<float2*>(grp_final + (size_t)g * D_DIM + d) = outv;
}

// ---------------------------------------------------------------------------
// Gather the 6 output sections (each B x 64), concatenated flat.
// ---------------------------------------------------------------------------
__global__ void gather_kernel(const int* __restrict__ ui, const int* __restrict__ pg,
                              const int* __restrict__ ng,
                              const float* __restrict__ hg_acc,
                              const float* __restrict__ grp_final,
                              const float* __restrict__ user_emb,
                              const float* __restrict__ group_emb,
                              float* __restrict__ out) {
  const size_t sec = (size_t)B_CNT * D_DIM;
  size_t t = (size_t)blockIdx.x * blockDim.x + threadIdx.x;
  if (t >= 6 * sec) return;
  int    s = (int)(t / sec);
  size_t r = (t % sec) / D_DIM;
  int    d = (int)(t % D_DIM);
  float  val;
  switch (s) {
    case 0:  val = hg_acc   [(size_t)ui[r] * D_DIM + d] * 0.25f; break;
    case 1:  val = grp_final[(size_t)pg[r] * D_DIM + d];         break;
    case 2:  val = grp_final[(size_t)ng[r] * D_DIM + d];         break;
    case 3:  val = user_emb [(size_t)ui[r] * D_DIM + d];         break;
    case 4:  val = group_emb[(size_t)pg[r] * D_DIM + d];         break;
    default: val = group_emb[(size_t)ng[r] * D_DIM + d];         break;
  }
  out[t] = val;
}

// ---------------------------------------------------------------------------
// Orchestration
// ---------------------------------------------------------------------------
extern "C" void kernel_launch(void* const* d_in, const int* in_sizes, int n_in,
                              void* d_out, int out_size, void* d_ws, size_t ws_size,
                              hipStream_t stream) {
  (void)in_sizes; (void)n_in; (void)out_size; (void)ws_size;

  const int*   user_inputs = (const int*)  d_in[0];
  const int*   pos_groups  = (const int*)  d_in[1];
  const int*   neg_groups  = (const int*)  d_in[2];
  const int*   hg_rows     = (const int*)  d_in[3];
  const int*   hg_cols     = (const int*)  d_in[4];
  const float* hg_vals     = (const float*)d_in[5];
  const int*   gi_rows     = (const int*)  d_in[6];
  const int*   gi_cols     = (const int*)  d_in[7];
  const float* gi_vals     = (const float*)d_in[8];
  const float* gg_graph    = (const float*)d_in[9];
  const float* user_emb    = (const float*)d_in[10];
  const float* item_emb    = (const float*)d_in[11];
  const float* group_emb   = (const float*)d_in[12];
  const float* hyper_w     = (const float*)d_in[13];
  const float* hyper_b     = (const float*)d_in[14];
  const float* lightgcn_w  = (const float*)d_in[15];
  const float* lightgcn_b  = (const float*)d_in[16];
  const float* overlap_w   = (const float*)d_in[17];
  const float* overlap_b   = (const float*)d_in[18];

  const size_t NHG = (size_t)NHG_ROWS * D_DIM;   // 13,440,000
  const size_t NGI = (size_t)NGI_ROWS * D_DIM;   //  7,040,000
  const size_t NG  = (size_t)G_CNT    * D_DIM;   //    640,000

  float* ws        = (float*)d_ws;
  float* hg_a      = ws;
  float* hg_b      = hg_a   + NHG;
  float* hg_acc    = hg_b   + NHG;
  float* gi_in     = hg_acc + NHG;
  float* gi_out    = gi_in  + NGI;
  float* gg_out    = gi_out + NGI;
  float* grp_final = gg_out + NG;

  const int TB = 256;

  // hg_cur = hg_acc = [user_emb ; group_emb]
  {
    int blocks = (int)((NHG + TB - 1) / TB);
    concat2_kernel<<<blocks, TB, 0, stream>>>(hg_a, hg_acc,
                                              user_emb, (size_t)U_CNT * D_DIM,
                                              group_emb, (size_t)G_CNT * D_DIM);
  }
  // gi_in = [group_emb ; item_emb]
  {
    int blocks = (int)((NGI + TB - 1) / TB);
    concat2_kernel<<<blocks, TB, 0, stream>>>(gi_in, (float*)nullptr,
                                              group_emb, (size_t)G_CNT * D_DIM,
                                              item_emb, (size_t)I_CNT * D_DIM);
  }
  hipMemsetAsync(gi_out, 0, NGI * sizeof(float), stream);
  hipMemsetAsync(gg_out, 0, NG  * sizeof(float), stream);

  // 3 hypergraph propagation layers with accumulation
  float* cur = hg_a;
  float* nxt = hg_b;
  const int spmmBlocks = 4096;
  for (int layer = 0; layer < 3; ++layer) {
    hipMemsetAsync(nxt, 0, NHG * sizeof(float), stream);
    spmm_atomic_kernel<<<spmmBlocks, TB, 0, stream>>>(hg_rows, hg_cols, hg_vals,
                                                      cur, nxt, EHG_CNT);
    int blocks = (int)((NHG + TB - 1) / TB);
    add_inplace_kernel<<<blocks, TB, 0, stream>>>(hg_acc, nxt, NHG);
    float* tmp = cur; cur = nxt; nxt = tmp;
  }

  // one GI propagation layer
  spmm_atomic_kernel<<<spmmBlocks, TB, 0, stream>>>(gi_rows, gi_cols, gi_vals,
                                                    gi_in, gi_out, EGI_CNT);

  // dense gg = gg_graph @ group_emb via f32 WMMA, K-split for bandwidth
  {
    int slabs  = G_CNT / 16;                     // 625
    int wpb    = TB / 32;                        // 8 waves per block
    dim3 grid((slabs + wpb - 1) / wpb,           // 79 slab-groups
              NKCHUNKS);                         // 16 K-chunks => 10000 waves
    gg_wmma_kernel<<<grid, TB, 0, stream>>>(gg_graph, group_emb, gg_out);
  }

  // gates + combine
  {
    int wpb    = TB / 32;
    int blocks = (G_CNT + wpb - 1) / wpb;        // 1250
    coef_combine_kernel<<<blocks, TB, 0, stream>>>(hg_acc, gi_out, gg_out,
                                                   hyper_w, hyper_b,
                                                   lightgcn_w, lightgcn_b,
                                                   overlap_w, overlap_b,
                                                   grp_final);
  }

  // final gathers into d_out (6 sections of B x 64)
  {
    size_t total  = 6ull * B_CNT * D_DIM;        // 3,145,728
    int    blocks = (int)((total + TB - 1) / TB);
    gather_kernel<<<blocks, TB, 0, stream>>>(user_inputs, pos_groups, neg_groups,
                                             hg_acc, grp_final, user_emb, group_emb,
                                             (float*)d_out);
  }
}